// RoIBBox_76759655514851
// MI455X (gfx1250) — compile-verified
//
#include <hip/hip_runtime.h>
#include <stdint.h>

#define NANC 8649      // 31*31*9 anchors
#define NT   6000      // PRE_NMS
#define POST 1500      // POST_NMS
#define NBATCH 16
#define IOU_T 0.7f
#define SORT_M 16384   // next pow2 >= NANC

typedef __attribute__((ext_vector_type(2))) float v2f;
typedef __attribute__((ext_vector_type(8))) float v8f;

// ---------------------------------------------------------------- zero output
__global__ void zero_kernel(float* __restrict__ out, int n) {
    int t = blockIdx.x * blockDim.x + threadIdx.x;
    if (t < n) out[t] = 0.0f;
}

// ------------------------------------------------------- decode anchors+deltas
__global__ void decode_kernel(const float* __restrict__ deltas,
                              const float* __restrict__ anchors,
                              float4* __restrict__ boxesWS, int total) {
    int t = blockIdx.x * blockDim.x + threadIdx.x;
    if (t >= total) return;
    int n = t % NANC;
    const float4 dp = ((const float4*)deltas)[t];
    const float4 ap = ((const float4*)anchors)[n];
    float dy = dp.x * 0.1f, dx = dp.y * 0.1f;
    float dh = dp.z * 0.2f, dw = dp.w * 0.2f;
    float ah = ap.z - ap.x, aw = ap.w - ap.y;
    float bh = __expf(dh) * ah;
    float bw = __expf(dw) * aw;
    float cy = ap.x + 0.5f * ah + dy * ah;
    float cx = ap.y + 0.5f * aw + dx * aw;
    float y1 = cy - 0.5f * bh, x1 = cx - 0.5f * bw;
    boxesWS[t] = make_float4(y1, x1, y1 + bh, x1 + bw);
}

// -------------------------------------------- per-batch bitonic top-k (in LDS)
// sortedWS per batch: [ boxes AoS float4 x NT | scores float x NT ] = 5*NT floats
extern "C" __global__ __launch_bounds__(1024)
void sort_kernel(const float* __restrict__ probs,
                 const float4* __restrict__ boxesWS,
                 float* __restrict__ sortedWS) {
    const int b = blockIdx.x;
    const int tid = threadIdx.x;
    extern __shared__ unsigned long long keys[];   // SORT_M * 8B = 128KB LDS

    for (int m = tid; m < SORT_M; m += 1024) {
        unsigned long long key = 0ull;
        if (m < NANC) {
            float s = probs[(size_t)b * NANC + m];
            unsigned bits = __float_as_uint(s);
            unsigned ord  = (bits & 0x80000000u) ? ~bits : (bits | 0x80000000u);
            key = ((unsigned long long)ord << 32) | (unsigned)(~(unsigned)m); // tie: low idx first
        }
        keys[m] = key;
    }
    __syncthreads();

    for (int k = 2; k <= SORT_M; k <<= 1) {
        for (int j = k >> 1; j > 0; j >>= 1) {
            for (int i = tid; i < SORT_M; i += 1024) {
                int l = i ^ j;
                if (l > i) {
                    unsigned long long a = keys[i], c = keys[l];
                    bool up = ((i & k) == 0);            // descending
                    if ((a < c) == up) { keys[i] = c; keys[l] = a; }
                }
            }
            __syncthreads();
        }
    }

    float4* dstB = (float4*)(sortedWS + (size_t)b * 5 * NT);
    float*  dstS = sortedWS + (size_t)b * 5 * NT + 4 * NT;
    for (int s = tid; s < NT; s += 1024) {
        unsigned long long key = keys[s];
        unsigned m   = ~((unsigned)key);
        unsigned ord = (unsigned)(key >> 32);
        unsigned bits = (ord & 0x80000000u) ? (ord & 0x7fffffffu) : ~ord;
        dstB[s] = boxesWS[(size_t)b * NANC + m];
        dstS[s] = __uint_as_float(bits);
    }
}

// ---------------------------------------------------- per-batch block-leader NMS
// LDS: boxes AoS float4[NT] + area f32[NT] + slot i32[NT] + supp bits + cnt + amask
#define NMS_LDS_BYTES (NT * 16 + NT * 4 + NT * 4 + 192 * 4 + 32)

extern "C" __global__ __launch_bounds__(512)
void nms_kernel(const float* __restrict__ sortedWS,
                float* __restrict__ outBoxes,
                float* __restrict__ outScores) {
    const int b = blockIdx.x;
    const int tid = threadIdx.x;
    const int lane = tid & 31;
    const int wid  = tid >> 5;            // 16 wave32s
    const float* base = sortedWS + (size_t)b * 5 * NT;

    extern __shared__ char smem_raw[];
    float4* bx  = (float4*)smem_raw;            // NT boxes (y1,x1,y2,x2)
    float*  ar  = (float*)(bx + NT);            // NT areas
    int*    slot = (int*)(ar + NT);             // NT output slots (-1 = dropped)
    unsigned* supp = (unsigned*)(slot + NT);    // 188 words used, 192 reserved
    int* cntp = (int*)(supp + 192);
    unsigned* amaskp = (unsigned*)(cntp + 1);   // active-leader mask for this block

    // ---- async-copy the AoS boxes (NT x float4 = 6000 x B128) into LDS ----
    {
        const uint32_t lds0 = (uint32_t)(uintptr_t)(void*)bx;
        for (uint32_t k = tid; k < (uint32_t)NT; k += blockDim.x) {
            uint32_t boff = k * 16u;
            asm volatile("global_load_async_to_lds_b128 %0, %1, %2"
                         :: "v"(lds0 + boff), "v"(boff), "s"(base)
                         : "memory");
        }
        asm volatile("s_wait_asynccnt 0" ::: "memory");
    }
    __syncthreads();

    for (int i = tid; i < NT; i += blockDim.x) {
        float4 bi = bx[i];
        ar[i] = (bi.z - bi.x) * (bi.w - bi.y);
        slot[i] = -1;
    }
    for (int i = tid; i < 192; i += blockDim.x) supp[i] = 0u;
    if (tid == 0) *cntp = 0;
    __syncthreads();

    const unsigned lane15 = (unsigned)(lane & 15);
    const bool loHalf = (lane < 16);

    // ---- greedy NMS over blocks of 16 leaders ----
    for (int i0 = 0; i0 < NT; i0 += 16) {
        // phase 1: sequential in-block resolution (single thread, <=120 IoUs)
        if (tid == 0) {
            int c = *cntp;
            unsigned am = 0u;
            for (int r = 0; r < 16; ++r) {
                int i = i0 + r;
                bool sup = (supp[i >> 5] >> (i & 31)) & 1u;
                if (!sup && c < POST) {
                    slot[i] = c++;
                    am |= (1u << r);
                    float4 bi = bx[i];
                    float aa = ar[i];
                    for (int r2 = r + 1; r2 < 16; ++r2) {
                        int j = i0 + r2;
                        float4 bj = bx[j];
                        float ih = fmaxf(fminf(bi.z, bj.z) - fmaxf(bi.x, bj.x), 0.0f);
                        float iw = fmaxf(fminf(bi.w, bj.w) - fmaxf(bi.y, bj.y), 0.0f);
                        float inter = ih * iw;
                        // iou > thr  <=>  inter > thr*(a_i+a_j-inter+eps)   (denom > 0)
                        if (inter > IOU_T * (aa + ar[j] - inter + 1e-8f))
                            supp[j >> 5] |= (1u << (j & 31));
                    }
                }
            }
            *cntp = c;
            *amaskp = am;
        }
        __syncthreads();

        const unsigned activeMask = *amaskp;      // uniform

        // phase 2: parallel 16x16 suppression tiles over all downstream boxes
        if (activeMask) {
            // A fragment (16x4 f32): col0 = area_i, col1 = 1  ->  D = a_i + a_j
            float aival = ar[i0 + lane15];                 // in-bounds for all lanes
            v2f afrag;
            afrag.x = loHalf ? aival : 0.0f;               // K=0 (low half-wave)
            afrag.y = loHalf ? 1.0f : 0.0f;                // K=1

            // preload this half-wave's 8 leader-row boxes into registers
            const int rbase = (lane >> 4) << 3;            // 0 or 8
            const unsigned actBits = (activeMask >> rbase) & 0xffu; // per-lane const
            float4 rb[8];
            float  rowBias[8];                             // 0 = active, -1e30 = masked
#pragma unroll
            for (int r = 0; r < 8; ++r) {
                rb[r] = bx[i0 + rbase + r];
                rowBias[r] = ((actBits >> r) & 1u) ? 0.0f : -1e30f;
            }

            const int ntiles = (NT - (i0 + 16)) >> 4;
            for (int t = wid; t < ntiles; t += 16) {
                const int j0  = i0 + 16 + (t << 4);
                const int col = j0 + (int)lane15;
                float ajval = ar[col];                     // in-bounds for all lanes
                v2f bfrag;
                bfrag.x = loHalf ? 1.0f : 0.0f;            // K=0 row of ones
                bfrag.y = loHalf ? ajval : 0.0f;           // K=1 row = area_j
                v8f cfr = {};
                v8f d = __builtin_amdgcn_wmma_f32_16x16x4_f32(
                    false, afrag, false, bfrag, (short)0, cfr, false, false);

                const float4 bj = bx[col];                 // one ds_load_b128
                float acc = -1.0f;                         // max suppression margin
#pragma unroll
                for (int r = 0; r < 8; ++r) {
                    float ih = fmaxf(fminf(rb[r].z, bj.z) - fmaxf(rb[r].x, bj.x), 0.0f);
                    float iw = fmaxf(fminf(rb[r].w, bj.w) - fmaxf(rb[r].y, bj.y), 0.0f);
                    float inter = ih * iw;
                    // margin > 0  <=>  iou > thr ; d[r] = a_i + a_j (from WMMA)
                    float denom  = d[r] - inter + 1e-8f;
                    float margin = fmaf(-IOU_T, denom, inter) + rowBias[r];
                    acc = fmaxf(acc, margin);
                }
                if (acc > 0.0f) atomicOr(&supp[col >> 5], 1u << (col & 31));
            }
        }
        __syncthreads();
    }

    // ---- scatter survivors (output pre-zeroed by zero_kernel) ----
    const float* sc = base + 4 * NT;
    for (int i = tid; i < NT; i += blockDim.x) {
        int s = slot[i];
        if (s >= 0) {
            float4 bi = bx[i];
            float* ob = outBoxes + ((size_t)b * POST + s) * 4;
            ob[0] = fminf(fmaxf(bi.x, 0.0f), 1.0f);
            ob[1] = fminf(fmaxf(bi.y, 0.0f), 1.0f);
            ob[2] = fminf(fmaxf(bi.z, 0.0f), 1.0f);
            ob[3] = fminf(fmaxf(bi.w, 0.0f), 1.0f);
            outScores[(size_t)b * POST + s] = sc[i];
        }
    }
}

// ---------------------------------------------------------------------- launch
extern "C" void kernel_launch(void* const* d_in, const int* in_sizes, int n_in,
                              void* d_out, int out_size, void* d_ws, size_t ws_size,
                              hipStream_t stream) {
    const float* deltas  = (const float*)d_in[0];   // [16,31,31,36]
    const float* probs   = (const float*)d_in[1];   // [16,31,31,9]
    const float* anchors = (const float*)d_in[2];   // [8649,4]
    float* out = (float*)d_out;                     // boxes[16*1500*4] ++ scores[16*1500]

    float4* boxesWS  = (float4*)d_ws;                               // B*NANC float4
    float*  sortedWS = (float*)d_ws + (size_t)NBATCH * NANC * 4;    // B*5*NT floats

    zero_kernel<<<(out_size + 255) / 256, 256, 0, stream>>>(out, out_size);

    const int total = NBATCH * NANC;
    decode_kernel<<<(total + 255) / 256, 256, 0, stream>>>(deltas, anchors, boxesWS, total);

    sort_kernel<<<NBATCH, 1024, SORT_M * sizeof(unsigned long long), stream>>>(
        probs, boxesWS, sortedWS);

    nms_kernel<<<NBATCH, 512, NMS_LDS_BYTES, stream>>>(
        sortedWS, out, out + (size_t)NBATCH * POST * 4);
}